// HdcRbfEncoder_3161095930405
// MI455X (gfx1250) — compile-verified
//
#include <hip/hip_runtime.h>
#include <math.h>

typedef __attribute__((ext_vector_type(2))) float v2f;
typedef __attribute__((ext_vector_type(8))) float v8f;

#define T_LEN 400
#define D_LEN 10000
#define K1 1200   // 3*T
#define K3 800    // 2*T
#define NTILES (D_LEN / 16)   // 625

// ---------------------------------------------------------------------------
// Kernel 1: feature extraction (tiny, one block).
// ws layout (floats): [0,1200) f1 = [x|y|z], [1200,2400) f2 = jerks [jx|jy|jz],
//                     [2400,3200) f3 = [mags|jerk_mags], [3200,3203) energy
// ---------------------------------------------------------------------------
__global__ void hdc_features_kernel(const float* __restrict__ input,
                                    float* __restrict__ ws) {
  __shared__ float s_t[T_LEN];
  __shared__ float s_x[T_LEN * 3];

  const int tid = threadIdx.x;
  for (int t = tid; t < T_LEN; t += blockDim.x) {
    s_t[t]         = input[t * 4 + 0];
    s_x[t * 3 + 0] = input[t * 4 + 1];
    s_x[t * 3 + 1] = input[t * 4 + 2];
    s_x[t * 3 + 2] = input[t * 4 + 3];
  }
  __syncthreads();

  for (int t = tid; t < T_LEN; t += blockDim.x) {
    const float x = s_x[t * 3 + 0];
    const float y = s_x[t * 3 + 1];
    const float z = s_x[t * 3 + 2];
    // features1: column-concatenated xyz
    ws[0 * T_LEN + t] = x;
    ws[1 * T_LEN + t] = y;
    ws[2 * T_LEN + t] = z;
    // jerks (row 0 is zero)
    float jx = 0.f, jy = 0.f, jz = 0.f;
    if (t > 0) {
      const float dt = s_t[t] - s_t[t - 1];
      const float inv = 1.0f / dt;
      jx = (x - s_x[(t - 1) * 3 + 0]) * inv;
      jy = (y - s_x[(t - 1) * 3 + 1]) * inv;
      jz = (z - s_x[(t - 1) * 3 + 2]) * inv;
    }
    ws[K1 + 0 * T_LEN + t] = jx;
    ws[K1 + 1 * T_LEN + t] = jy;
    ws[K1 + 2 * T_LEN + t] = jz;
    // features3: [mags | jerk_mags]
    ws[2 * K1 + t]         = sqrtf(x * x + y * y + z * z);
    ws[2 * K1 + T_LEN + t] = sqrtf(jx * jx + jy * jy + jz * jz);
  }
  __syncthreads();

  if (tid < 3) {   // energy[c] = sum_t xyz[t][c]^2 / T
    float s = 0.f;
    for (int t = 0; t < T_LEN; ++t) {
      const float v = s_x[t * 3 + tid];
      s += v * v;
    }
    ws[2 * K1 + K3 + tid] = s / (float)T_LEN;
  }
}

// ---------------------------------------------------------------------------
// Kernel 2: bandwidth-bound GEMV via V_WMMA_F32_16X16X4_F32.
// One wave per 16-row tile. A-fragment = weight tile (1x b64/lane/step),
// B-fragment = feature vector broadcast across all 16 columns.
// ---------------------------------------------------------------------------
__global__ __launch_bounds__(256) void hdc_gemv_wmma_kernel(
    const float* __restrict__ W1, const float* __restrict__ b1,
    const float* __restrict__ W2, const float* __restrict__ b2,
    const float* __restrict__ W3, const float* __restrict__ b3,
    const float* __restrict__ W4, const float* __restrict__ b4,
    const float* __restrict__ feats, float* __restrict__ out) {
  const int lane  = threadIdx.x & 31;
  const int wave  = threadIdx.x >> 5;
  const int tile  = blockIdx.x * (blockDim.x >> 5) + wave;
  if (tile >= NTILES) return;            // wave-uniform guard (EXEC stays all-1s)

  const int row0  = tile * 16;
  const int m     = lane & 15;           // A-matrix row this lane supplies
  const int khalf = (lane >> 4) * 2;     // 0 for lanes 0-15, 2 for lanes 16-31
  const int rowA  = row0 + m;

  const float* __restrict__ f1 = feats;
  const float* __restrict__ f2 = feats + K1;
  const float* __restrict__ f3 = feats + 2 * K1;
  const float* __restrict__ en = feats + 2 * K1 + K3;

  const float* __restrict__ r1 = W1 + (size_t)rowA * K1 + khalf;
  const float* __restrict__ r2 = W2 + (size_t)rowA * K1 + khalf;
  const float* __restrict__ r3 = W3 + (size_t)rowA * K3 + khalf;

  // Warm the row streams (lowers to global_prefetch_b8).
  __builtin_prefetch(r1, 0, 0);
  __builtin_prefetch(r2, 0, 0);
  __builtin_prefetch(r3, 0, 0);

  v8f acc1 = {};
  v8f acc2 = {};
  v8f acc3 = {};

  // Interleave the two K=1200 streams to keep two independent WMMA chains.
  #pragma unroll 4
  for (int k = 0; k < K1; k += 4) {
    const v2f a1 = *(const v2f*)(r1 + k);
    const v2f a2 = *(const v2f*)(r2 + k);
    const v2f bA = *(const v2f*)(f1 + k + khalf);
    const v2f bB = *(const v2f*)(f2 + k + khalf);
    acc1 = __builtin_amdgcn_wmma_f32_16x16x4_f32(false, a1, false, bA,
                                                 (short)0, acc1, false, false);
    acc2 = __builtin_amdgcn_wmma_f32_16x16x4_f32(false, a2, false, bB,
                                                 (short)0, acc2, false, false);
  }

  #pragma unroll 4
  for (int k = 0; k < K3; k += 4) {
    const v2f a3 = *(const v2f*)(r3 + k);
    const v2f bC = *(const v2f*)(f3 + k + khalf);
    acc3 = __builtin_amdgcn_wmma_f32_16x16x4_f32(false, a3, false, bC,
                                                 (short)0, acc3, false, false);
  }

  // C/D layout: VGPR v, lane half h -> row = row0 + 8*h + v; all 16 columns
  // carry the same value (broadcast B), so one lane per half finalizes 8 rows.
  if ((lane & 15) == 0) {
    const int   half = lane >> 4;
    const float e0 = en[0], e1 = en[1], e2 = en[2];
    #pragma unroll
    for (int v = 0; v < 8; ++v) {
      const int row = row0 + half * 8 + v;
      const float h1 = tanhf(acc1[v] + b1[row]);
      const float h2 = tanhf(acc2[v] + b2[row]);
      const float h3 = tanhf(acc3[v] + b3[row]);
      const float d4 = W4[row * 3 + 0] * e0 + W4[row * 3 + 1] * e1 +
                       W4[row * 3 + 2] * e2 + b4[row];
      const float h4 = tanhf(d4);
      out[row] = tanhf(h1 * h4 + h2 + h3);
    }
  }
}

// ---------------------------------------------------------------------------
extern "C" void kernel_launch(void* const* d_in, const int* in_sizes, int n_in,
                              void* d_out, int out_size, void* d_ws, size_t ws_size,
                              hipStream_t stream) {
  (void)in_sizes; (void)n_in; (void)out_size; (void)ws_size;
  const float* input = (const float*)d_in[0];
  const float* W1    = (const float*)d_in[1];
  const float* b1    = (const float*)d_in[2];
  const float* W2    = (const float*)d_in[3];
  const float* b2    = (const float*)d_in[4];
  const float* W3    = (const float*)d_in[5];
  const float* b3    = (const float*)d_in[6];
  const float* W4    = (const float*)d_in[7];
  const float* b4    = (const float*)d_in[8];
  float*       out   = (float*)d_out;
  float*       feats = (float*)d_ws;   // 3203 floats used

  hdc_features_kernel<<<1, 512, 0, stream>>>(input, feats);

  const int wavesPerBlock = 8;               // 256 threads
  const int blocks = (NTILES + wavesPerBlock - 1) / wavesPerBlock;  // 79
  hdc_gemv_wmma_kernel<<<blocks, 256, 0, stream>>>(W1, b1, W2, b2, W3, b3,
                                                   W4, b4, feats, out);
}